// GraphAttentionGroup_45732811767831
// MI455X (gfx1250) — compile-verified
//
#include <hip/hip_runtime.h>
#include <math.h>

#define N_NODES 10000
#define KNN     32
#define DG      256   // per-group feature dim (= OUT_DIM)

typedef __attribute__((ext_vector_type(2))) float v2f;
typedef __attribute__((ext_vector_type(8))) float v8f;

// ---------------------------------------------------------------------------
// Kernel 1: fw[g] = softmax(a_g) over 256 features. grid=2 blocks of 256.
// ---------------------------------------------------------------------------
__global__ __launch_bounds__(256) void softmax_a_kernel(const float* __restrict__ a1,
                                                        const float* __restrict__ a2,
                                                        float* __restrict__ fw) {
    const int g = blockIdx.x;
    const float* a = g ? a2 : a1;
    const int t = threadIdx.x;
    __shared__ float red[256];

    float v = a[t];
    red[t] = v;
    __syncthreads();
    for (int s = 128; s > 0; s >>= 1) {
        if (t < s) red[t] = fmaxf(red[t], red[t + s]);
        __syncthreads();
    }
    float mx = red[0];
    __syncthreads();
    float e = expf(v - mx);
    red[t] = e;
    __syncthreads();
    for (int s = 128; s > 0; s >>= 1) {
        if (t < s) red[t] += red[t + s];
        __syncthreads();
    }
    fw[g * DG + t] = e / red[0];
}

// ---------------------------------------------------------------------------
// Kernel 2: H[g][n][d] = clip(fw[g][d] * (x_g @ W_g^T)[n][d], -1, 1)
// Full-precision f32 WMMA: V_WMMA_F32_16X16X4_F32, K-loop of 64 steps.
// grid = (625 m-tiles, 4 col-quads, 2 groups), block = 128 (4 waves),
// each wave computes one 16x16 output tile.
//
// VGPR layouts (ISA 7.12.2):
//   A 16x4:  v0 = K0(l<16)/K2(l>=16), v1 = K1/K3 ; row M = lane&15
//   B 4x16:  v0 = Krow0(l<16)/Krow2(l>=16), v1 = Krow1/Krow3 ; col N = lane&15
//   C/D:     vj = row j (lanes 0-15) / row j+8 (lanes 16-31), col N = lane&15
// ---------------------------------------------------------------------------
__global__ __launch_bounds__(128) void gemm_wmma_kernel(const float* __restrict__ x,
                                                        const float* __restrict__ W1,
                                                        const float* __restrict__ W2,
                                                        const float* __restrict__ fw,
                                                        float* __restrict__ H) {
    const int lane = threadIdx.x & 31;
    const int wave = threadIdx.x >> 5;
    const int g    = blockIdx.z;

    const float* W  = g ? W2 : W1;
    const float* xg = x + g * DG;           // x row stride = 2*DG

    const int m0   = blockIdx.x * 16;
    const int n0   = blockIdx.y * 64 + wave * 16;
    const int half = lane >> 4;             // 0 or 1
    const int l15  = lane & 15;
    const int koff = half * 2;              // K sub-offset: 0 or 2

    const int arow = m0 + l15;              // A: M = lane&15
    const int bcol = n0 + l15;              // B: N = lane&15

    const float* ap = xg + (size_t)arow * (2 * DG) + koff;
    const float* bp = W  + (size_t)bcol * DG + koff;   // B[k][n] = W[n][k]

    v8f c = {0.f, 0.f, 0.f, 0.f, 0.f, 0.f, 0.f, 0.f};
#pragma unroll 4
    for (int kb = 0; kb < DG; kb += 4) {
        v2f av = *(const v2f*)(ap + kb);    // K = kb+koff, kb+koff+1
        v2f bv = *(const v2f*)(bp + kb);
        c = __builtin_amdgcn_wmma_f32_16x16x4_f32(false, av, false, bv,
                                                  (short)0, c, false, false);
    }

    const float f = fw[g * DG + bcol];
    float* Hg = H + ((size_t)g * N_NODES + m0) * DG;
#pragma unroll
    for (int j = 0; j < 8; ++j) {
        float h = c[j] * f;
        h = fminf(fmaxf(h, -1.f), 1.f);     // Hardtanh
        Hg[(size_t)(j + half * 8) * DG + bcol] = h;
    }
}

// ---------------------------------------------------------------------------
// Kernel 3: per-(node,group) gaussian attention over KNN=32 neighbors.
// One wave32 per (node,group); lane owns 8 features + 1 neighbor index.
// Pass 1: scores via squared distance + wave reduction; wave softmax.
// Pass 2: re-gather (L2-resident) weighted aggregation.
// ---------------------------------------------------------------------------
__global__ __launch_bounds__(256) void gat_att_kernel(const float* __restrict__ H,
                                                      const int* __restrict__ graph,
                                                      float* __restrict__ out) {
    const int lane = threadIdx.x & 31;
    const int wid  = (int)((blockIdx.x * blockDim.x + threadIdx.x) >> 5);
    if (wid >= 2 * N_NODES) return;
    const int g = wid & 1;
    const int n = wid >> 1;

    const float* Hg = H + (size_t)g * N_NODES * DG;

    // own features: lane handles dims [lane*8, lane*8+8)
    const float4* orow = (const float4*)(Hg + (size_t)n * DG) + lane * 2;
    const float4 o0 = orow[0];
    const float4 o1 = orow[1];

    const int myidx = graph[n * KNN + lane];   // this lane's neighbor id

    // ---- pass 1: attention scores ----
    float myscore = 0.f;
    for (int k = 0; k < KNN; ++k) {
        const int j = __shfl(myidx, k, 32);
        const float4* nr = (const float4*)(Hg + (size_t)j * DG) + lane * 2;
        const float4 b0 = nr[0];
        const float4 b1 = nr[1];
        float d2 = 0.f;
        float t;
        t = o0.x - b0.x; d2 += t * t;
        t = o0.y - b0.y; d2 += t * t;
        t = o0.z - b0.z; d2 += t * t;
        t = o0.w - b0.w; d2 += t * t;
        t = o1.x - b1.x; d2 += t * t;
        t = o1.y - b1.y; d2 += t * t;
        t = o1.z - b1.z; d2 += t * t;
        t = o1.w - b1.w; d2 += t * t;
#pragma unroll
        for (int off = 16; off > 0; off >>= 1) d2 += __shfl_xor(d2, off, 32);
        if (lane == k) myscore = -d2;
    }

    // ---- wave32 softmax over the 32 scores (one per lane) ----
    float m = myscore;
#pragma unroll
    for (int off = 16; off > 0; off >>= 1) m = fmaxf(m, __shfl_xor(m, off, 32));
    const float e = expf(myscore - m);
    float s = e;
#pragma unroll
    for (int off = 16; off > 0; off >>= 1) s += __shfl_xor(s, off, 32);
    const float att = e / s;

    // ---- pass 2: weighted aggregation ----
    float4 acc0 = {0.f, 0.f, 0.f, 0.f};
    float4 acc1 = {0.f, 0.f, 0.f, 0.f};
    for (int k = 0; k < KNN; ++k) {
        const float w = __shfl(att, k, 32);
        const int   j = __shfl(myidx, k, 32);
        const float4* nr = (const float4*)(Hg + (size_t)j * DG) + lane * 2;
        const float4 b0 = nr[0];
        const float4 b1 = nr[1];
        acc0.x += w * b0.x; acc0.y += w * b0.y;
        acc0.z += w * b0.z; acc0.w += w * b0.w;
        acc1.x += w * b1.x; acc1.y += w * b1.y;
        acc1.z += w * b1.z; acc1.w += w * b1.w;
    }

    float* op = out + (size_t)n * (2 * DG) + g * DG + lane * 8;
    ((float4*)op)[0] = acc0;
    ((float4*)op)[1] = acc1;
}

// ---------------------------------------------------------------------------
extern "C" void kernel_launch(void* const* d_in, const int* in_sizes, int n_in,
                              void* d_out, int out_size, void* d_ws, size_t ws_size,
                              hipStream_t stream) {
    (void)in_sizes; (void)n_in; (void)out_size; (void)ws_size;

    const float* x     = (const float*)d_in[0];   // (N, 512)
    const float* W1    = (const float*)d_in[1];   // (256, 256)
    const float* a1    = (const float*)d_in[2];   // (256,)
    const float* W2    = (const float*)d_in[3];   // (256, 256)
    const float* a2    = (const float*)d_in[4];   // (256,)
    const int*   graph = (const int*)d_in[5];     // (N, 32)
    float* out = (float*)d_out;                   // (N, 512)

    float* fw = (float*)d_ws;                     // 2*256 floats
    float* H  = fw + 2 * DG;                      // 2*N*256 floats (~20.5 MB)

    softmax_a_kernel<<<2, 256, 0, stream>>>(a1, a2, fw);
    gemm_wmma_kernel<<<dim3(N_NODES / 16, 4, 2), 128, 0, stream>>>(x, W1, W2, fw, H);
    gat_att_kernel<<<(2 * N_NODES) / 8, 256, 0, stream>>>(H, graph, out);
}